// GCN_Layer_35184372089478
// MI455X (gfx1250) — compile-verified
//
#include <hip/hip_runtime.h>

#define ATOM_FEA 128
#define NBR_FEA  64
#define NUM_NBR  12
#define N_NODES  50000
#define N_EDGES  (N_NODES * NUM_NBR)   // 600000
#define K_DIM    320                   // 2*ATOM_FEA + NBR_FEA
#define OUT_DIM  256                   // 2*ATOM_FEA
#define BN_EPS   1e-5f

typedef __attribute__((ext_vector_type(2))) float v2f;
typedef __attribute__((ext_vector_type(8))) float v8f;
typedef __attribute__((__vector_size__(4 * sizeof(int)))) int v4i;   // async b128 payload type

#define MTILE 128      // edges per block
#define KCH   64       // K-chunk staged in LDS
#define LDA   68       // LDS pitch (floats): 68 mod 64 = 4 -> conflict-free b64 frag loads
#define ABUF  (MTILE * LDA)            // floats per A buffer

#if defined(__has_builtin)
#  if __has_builtin(__builtin_amdgcn_global_load_async_to_lds_b128)
#    define HAS_ASYNC 1
#  else
#    define HAS_ASYNC 0
#  endif
#else
#  define HAS_ASYNC 0
#endif

__device__ __forceinline__ float sigmoid_f(float v)  { return 1.0f / (1.0f + __expf(-v)); }
__device__ __forceinline__ float softplus_f(float v) { return v > 20.0f ? v : log1pf(__expf(v)); }

__device__ __forceinline__ void wait_async0() {
#if HAS_ASYNC
#  if __has_builtin(__builtin_amdgcn_s_wait_asynccnt)
    __builtin_amdgcn_s_wait_asynccnt(0);
#  else
    asm volatile("s_wait_asynccnt 0" ::: "memory");
#  endif
#endif
}

// ---------------------------------------------------------------------------
// Kernel 1: repack W[256,320] -> pair-interleaved Wt so a WMMA B-fragment
// (rows k..k+3 of W^T, 16 columns) is a single coalesced float2 load:
//   Wtp[(k>>1)*512 + 2*n + (k&1)] = W[n*320 + k]
// Also zero BN1 stat accumulators (512 floats).
// ---------------------------------------------------------------------------
__global__ void prep_kernel(const float* __restrict__ W, float* __restrict__ Wtp,
                            float* __restrict__ s1)
{
    const int idx = blockIdx.x * 256 + threadIdx.x;   // 0 .. 81919
    const int k = idx >> 8;
    const int n = idx & 255;
    Wtp[(k >> 1) * 512 + 2 * n + (k & 1)] = W[n * K_DIM + k];
    if (blockIdx.x == 0) {
        s1[threadIdx.x] = 0.0f;          // sum
        s1[256 + threadIdx.x] = 0.0f;    // sumsq
    }
}

// ---------------------------------------------------------------------------
// Gather-stage one [MTILE x KCH] chunk of the concatenated edge features into
// LDS. Uses CDNA5 async global->LDS (no VGPR round trip, ASYNCcnt-tracked)
// when available; issuing thread does NOT wait here (double buffering).
// ---------------------------------------------------------------------------
__device__ __forceinline__ void stage_chunk(
    float* __restrict__ lbuf, const int kc, const long long eBase, const int tid,
    const float* __restrict__ x, const long long* __restrict__ nbr,
    const float* __restrict__ ea)
{
    const int r   = tid >> 1;              // 2 threads per edge row
    const int seg = (tid & 1) * 32;
    const long long e = eBase + r;
    const bool valid = (e < (long long)N_EDGES);
    const float* srcp;
    if (kc < ATOM_FEA) {                                  // x[nbr[e]], cols kc..kc+63
        srcp = x + (valid ? nbr[e] : 0) * ATOM_FEA + kc;
    } else if (kc < 2 * ATOM_FEA) {                       // x[src[e]] = x[e/12]
        srcp = x + (valid ? (e / NUM_NBR) : 0) * ATOM_FEA + (kc - ATOM_FEA);
    } else {                                              // edge_attr[e]
        srcp = ea + (valid ? e : 0) * NBR_FEA;
    }
    float* lrow = lbuf + r * LDA + seg;
#pragma unroll
    for (int i = 0; i < 8; ++i) {
        if (valid) {
#if HAS_ASYNC
            __builtin_amdgcn_global_load_async_to_lds_b128(
                (v4i*)(srcp + seg + i * 4), (v4i*)(lrow + i * 4), 0, 0);
#else
            *(float4*)(lrow + i * 4) = *(const float4*)(srcp + seg + i * 4);
#endif
        } else {
            *(float4*)(lrow + i * 4) = make_float4(0.f, 0.f, 0.f, 0.f);
        }
    }
}

// ---------------------------------------------------------------------------
// Kernel 2: fused gather + GEMM (fp32 WMMA 16x16x4) + BN1 statistics.
// Block = 256 threads = 8 waves. Wave w owns M-strip [w*16, w*16+16) of a
// 128-edge tile, all 16 N-tiles (16 v8f accumulators). A chunks are
// double-buffered in LDS via async gather; B fragments stream from the
// L2-resident pair-interleaved Wtp.
// ---------------------------------------------------------------------------
__global__ __launch_bounds__(256) void gemm_kernel(
    const float* __restrict__ x, const long long* __restrict__ nbr,
    const float* __restrict__ ea, const float* __restrict__ Wtp,
    float* __restrict__ gated, float* __restrict__ s1sum, float* __restrict__ s1sq)
{
    extern __shared__ float ldsA[];       // 2 * ABUF floats (double buffer)
    __shared__ float lsum[OUT_DIM];
    __shared__ float lsq[OUT_DIM];

    const int tid  = threadIdx.x;
    const int wave = tid >> 5;
    const int lane = tid & 31;
    const int half = lane >> 4;           // 0: lanes 0-15, 1: lanes 16-31
    const int ml   = lane & 15;
    const long long eBase = (long long)blockIdx.x * MTILE;

    lsum[tid] = 0.0f;
    lsq[tid]  = 0.0f;

    v8f acc[16];
#pragma unroll
    for (int i = 0; i < 16; ++i) {
        v8f z = {0.f,0.f,0.f,0.f,0.f,0.f,0.f,0.f};
        acc[i] = z;
    }

    // prologue: stage chunk 0
    stage_chunk(ldsA, 0, eBase, tid, x, nbr, ea);
    wait_async0();
    __syncthreads();

    for (int ci = 0; ci < K_DIM / KCH; ++ci) {            // 5 chunks
        float* cur = ldsA + (ci & 1) * ABUF;
        if (ci < K_DIM / KCH - 1)                          // prefetch next chunk
            stage_chunk(ldsA + ((ci + 1) & 1) * ABUF, (ci + 1) * KCH, eBase, tid, x, nbr, ea);

        const int kc = ci * KCH;
#pragma unroll
        for (int ks = 0; ks < KCH / 4; ++ks) {
            const int k0 = ks * 4;
            // A fragment: VGPR0 = A[m][k0+2*half], VGPR1 = A[m][k0+1+2*half]
            v2f a = *(const v2f*)(&cur[(wave * 16 + ml) * LDA + k0 + 2 * half]);
            // B fragment base: pair (k0,k0+1) for lanes 0-15, (k0+2,k0+3) for 16-31
            const float* bbase = Wtp + (long long)(((kc + k0) >> 1) + half) * 512 + 2 * ml;
#pragma unroll
            for (int nt = 0; nt < 16; ++nt) {
                v2f b = *(const v2f*)(bbase + nt * 32);
                acc[nt] = __builtin_amdgcn_wmma_f32_16x16x4_f32(
                    false, a, false, b, (short)0, acc[nt], false, false);
            }
        }
        wait_async0();        // our async writes to the other buffer done
        __syncthreads();      // everyone's writes visible / reads finished
    }

    // ---- epilogue: store gated, accumulate per-channel sum / sumsq ----
    const long long eRow0 = eBase + wave * 16 + 8 * half;  // C layout: vgpr r -> M = r + 8*half
#pragma unroll
    for (int nt = 0; nt < 16; ++nt) {
        const int ch = nt * 16 + ml;
        float s = 0.0f, q = 0.0f;
#pragma unroll
        for (int r = 0; r < 8; ++r) {
            const float v = acc[nt][r];
            s += v;
            q += v * v;
            const long long e = eRow0 + r;
            if (e < (long long)N_EDGES) gated[e * OUT_DIM + ch] = v;
        }
        atomicAdd(&lsum[ch], s);    // ds float atomics
        atomicAdd(&lsq[ch],  q);
    }
    __syncthreads();
    atomicAdd(&s1sum[tid], lsum[tid]);
    atomicAdd(&s1sq[tid],  lsq[tid]);
}

// ---------------------------------------------------------------------------
// Kernel 3: fold BN1 stats into affine scale/shift; zero BN2 stats (256 floats)
// ---------------------------------------------------------------------------
__global__ void bn1_finalize_kernel(const float* __restrict__ s1sum, const float* __restrict__ s1sq,
                                    const float* __restrict__ g, const float* __restrict__ b,
                                    float* __restrict__ scale1, float* __restrict__ shift1,
                                    float* __restrict__ s2)
{
    const int c = threadIdx.x;                 // 256
    const float invE = 1.0f / (float)N_EDGES;
    const float mean = s1sum[c] * invE;
    const float var  = s1sq[c] * invE - mean * mean;
    const float sc   = g[c] * rsqrtf(var + BN_EPS);
    scale1[c] = sc;
    shift1[c] = b[c] - mean * sc;
    s2[c] = 0.0f;                              // zeroes s2sum(128) + s2sq(128)
}

// ---------------------------------------------------------------------------
// Kernel 4: BN1 affine + sigmoid*softplus + 12-neighbor sum + BN2 statistics.
// One wave per node, 8 nodes per block.
// ---------------------------------------------------------------------------
__global__ __launch_bounds__(256) void aggregate_kernel(
    const float* __restrict__ gated, const float* __restrict__ scale1,
    const float* __restrict__ shift1, float* __restrict__ nsum,
    float* __restrict__ s2sum, float* __restrict__ s2sq)
{
    __shared__ float lsum[ATOM_FEA];
    __shared__ float lsq[ATOM_FEA];
    const int tid = threadIdx.x;
    if (tid < ATOM_FEA) { lsum[tid] = 0.0f; lsq[tid] = 0.0f; }
    __syncthreads();

    const int wave = tid >> 5, lane = tid & 31;
    const long long node = (long long)blockIdx.x * 8 + wave;
    if (node < (long long)N_NODES) {
        for (int cb = 0; cb < ATOM_FEA; cb += 32) {
            const int c  = cb + lane;
            const float sf  = scale1[c],            bf  = shift1[c];
            const float scr = scale1[c + ATOM_FEA], bcr = shift1[c + ATOM_FEA];
            float accv = 0.0f;
            const float* rowp = gated + node * NUM_NBR * OUT_DIM;
#pragma unroll
            for (int j = 0; j < NUM_NBR; ++j) {
                const float gf = rowp[c] * sf + bf;                 // nbr_filter
                const float gc = rowp[c + ATOM_FEA] * scr + bcr;    // nbr_core
                accv += sigmoid_f(gf) * softplus_f(gc);
                rowp += OUT_DIM;
            }
            nsum[node * ATOM_FEA + c] = accv;
            atomicAdd(&lsum[c], accv);
            atomicAdd(&lsq[c],  accv * accv);
        }
    }
    __syncthreads();
    if (tid < ATOM_FEA) {
        atomicAdd(&s2sum[tid], lsum[tid]);
        atomicAdd(&s2sq[tid],  lsq[tid]);
    }
}

// ---------------------------------------------------------------------------
// Kernel 5: fold BN2 stats into affine scale/shift
// ---------------------------------------------------------------------------
__global__ void bn2_finalize_kernel(const float* __restrict__ s2sum, const float* __restrict__ s2sq,
                                    const float* __restrict__ g, const float* __restrict__ b,
                                    float* __restrict__ scale2, float* __restrict__ shift2)
{
    const int c = threadIdx.x;                 // 128
    const float invN = 1.0f / (float)N_NODES;
    const float mean = s2sum[c] * invN;
    const float var  = s2sq[c] * invN - mean * mean;
    const float sc   = g[c] * rsqrtf(var + BN_EPS);
    scale2[c] = sc;
    shift2[c] = b[c] - mean * sc;
}

// ---------------------------------------------------------------------------
// Kernel 6: out = softplus(x + BN2(nbr_sumed))
// ---------------------------------------------------------------------------
__global__ void output_kernel(const float* __restrict__ x, const float* __restrict__ nsum,
                              const float* __restrict__ scale2, const float* __restrict__ shift2,
                              float* __restrict__ out)
{
    const long long i = (long long)blockIdx.x * 256 + threadIdx.x;
    if (i < (long long)N_NODES * ATOM_FEA) {
        const int c = (int)(i & (ATOM_FEA - 1));
        const float h = nsum[i] * scale2[c] + shift2[c];
        out[i] = softplus_f(x[i] + h);
    }
}

// ---------------------------------------------------------------------------
extern "C" void kernel_launch(void* const* d_in, const int* in_sizes, int n_in,
                              void* d_out, int out_size, void* d_ws, size_t ws_size,
                              hipStream_t stream)
{
    const float*     x    = (const float*)d_in[0];
    const long long* eidx = (const long long*)d_in[1];  // int64 [2, E]
    const float*     ea   = (const float*)d_in[2];
    const float*     W    = (const float*)d_in[3];
    const float*     b    = (const float*)d_in[4];
    const float*     g1   = (const float*)d_in[5];
    const float*     b1   = (const float*)d_in[6];
    const float*     g2   = (const float*)d_in[7];
    const float*     b2   = (const float*)d_in[8];
    float* out = (float*)d_out;

    // workspace layout (floats)
    float* ws    = (float*)d_ws;
    float* Wtp   = ws;                        // 81920  (pair-interleaved W^T)
    float* s1sum = Wtp + 81920;               // 256
    float* s1sq  = s1sum + 256;               // 256    (contiguous with s1sum)
    float* sc1   = s1sq + 256;                // 256
    float* sh1   = sc1 + 256;                 // 256
    float* s2sum = sh1 + 256;                 // 128
    float* s2sq  = s2sum + 128;               // 128    (contiguous with s2sum)
    float* sc2   = s2sq + 128;                // 128
    float* sh2   = sc2 + 128;                 // 128
    float* gated = sh2 + 128;                 // E * 256
    float* nsum  = gated + (long long)N_EDGES * OUT_DIM;  // N * 128

    const long long* nbr = eidx + N_EDGES;    // edge_index[1] (neighbor ids)

    prep_kernel<<<(K_DIM * OUT_DIM) / 256, 256, 0, stream>>>(W, Wtp, s1sum);

    const int gblocks = (N_EDGES + MTILE - 1) / MTILE;   // 4688 (tail guarded)
    const size_t dynLds = 2 * ABUF * sizeof(float);      // 69632 B double buffer
    gemm_kernel<<<gblocks, 256, dynLds, stream>>>(x, nbr, ea, Wtp, gated, s1sum, s1sq);

    bn1_finalize_kernel<<<1, 256, 0, stream>>>(s1sum, s1sq, g1, b1, sc1, sh1, s2sum);

    aggregate_kernel<<<(N_NODES + 7) / 8, 256, 0, stream>>>(gated, sc1, sh1, nsum, s2sum, s2sq);

    bn2_finalize_kernel<<<1, 128, 0, stream>>>(s2sum, s2sq, g2, b2, sc2, sh2);

    output_kernel<<<(N_NODES * ATOM_FEA + 255) / 256, 256, 0, stream>>>(x, nsum, sc2, sh2, out);
}